// ModelSTAGIN_54288386621787
// MI455X (gfx1250) — compile-verified
//
#include <hip/hip_runtime.h>

// ---------------------------------------------------------------------------
// STAGIN forward for MI455X (gfx1250, wave32, WMMA).
// - All dense GEMMs run on V_WMMA_F32_16X16X4_F32 (f32-exact matrix pipe).
// - Memory-bound workload: each wave computes a 16x128 output strip so the
//   large A operand streams from HBM exactly once (B operand is L2-resident).
// - Tiles-per-wave is a template parameter: inner WMMA loop has a static trip
//   count -> no EXEC masking around WMMA, clean load/WMMA software pipelining.
// - GIN BN1+ReLU and MLP ReLU are fused into the consumer GEMM's A-loads.
// ---------------------------------------------------------------------------

typedef float v2f __attribute__((ext_vector_type(2)));
typedef float v8f __attribute__((ext_vector_type(8)));

static constexpr int Bb = 8, Tt = 64, Nn = 400, Cc = 400, Hh = 128, Ll = 2;
static constexpr int BT = Bb * Tt;                       // 512
static constexpr int ROWS_BTN = Bb * Tt * Nn;            // 204800
static constexpr int NSLICE = Nn * Nn;                   // 160000 per (b,t)

static constexpr long long O_LOGIT = 0;                                        // [B,2]
static constexpr long long O_NODE  = 16;                                       // [B,L,T,N]
static constexpr long long O_TIME  = O_NODE + (long long)Bb * Ll * Tt * Nn;    // 409616
static constexpr long long O_FG    = O_TIME + (long long)Bb * Ll * Tt * Tt;    // 475152
static constexpr long long O_FT    = O_FG   + (long long)Bb * Ll * Hh;         // 477200
static constexpr long long O_FL    = O_FT   + (long long)Bb * Ll * Hh;         // 479248

// ---------------------------------------------------------------------------
// WMMA helper: D = A(16x4) * B(4x16) + C, f32.
// A layout: lanes 0-15 => M=lane, v.x=K(2*half), v.y=K(2*half+1), half=lane>>4
// B layout: lanes (l = lane&15) => N=l, v.x=B[2*half][N], v.y=B[2*half+1][N]
// C/D: VGPR r: lanes0-15 -> M=r,N=lane ; lanes16-31 -> M=r+8,N=lane-16
// ---------------------------------------------------------------------------
static __device__ __forceinline__ v8f wmma4(v2f a, v2f b, v8f c) {
  return __builtin_amdgcn_wmma_f32_16x16x4_f32(false, a, false, b, (short)0, c,
                                               false, false);
}

// ---------------------------------------------------------------------------
// Strip GEMM: C[M,N] = actA(A[M,K]*aScale+aShift) @ W[K,N] (+ bias[N]).
// One wave owns a 16-row x (NT*16)-col strip: A read once, NT accumulators in
// VGPRs, static inner trip count. M%16==0, N%16==0, K%16==0.
// Column chunks: tn0 = tnBase + chunk*(NT*16), chunk in [0, nChunks).
// ---------------------------------------------------------------------------
template <int NT>
__global__ void k_gemm_strip(const float* __restrict__ A,
                             const float* __restrict__ W,
                             const float* __restrict__ bias,
                             const float* __restrict__ aScale,
                             const float* __restrict__ aShift,
                             int actA, float* __restrict__ C,
                             int M, int N, int K, int tnBase, int nChunks) {
  int wave = threadIdx.x >> 5;
  int lane = threadIdx.x & 31;
  int strips = (M >> 4) * nChunks;
  int strip = blockIdx.x * 8 + wave;
  if (strip >= strips) return;
  int rowTile = strip / nChunks;
  int chunk = strip - rowTile * nChunks;
  int tm = rowTile << 4;
  int tn0 = tnBase + chunk * (NT * 16);
  int half = lane >> 4, l = lane & 15;

  const float* arow = A + (size_t)(tm + l) * K + 2 * half;
  const float* brow = W + tn0 + l;

  v8f c[NT];
#pragma unroll
  for (int j = 0; j < NT; ++j) c[j] = (v8f){};

  for (int k0 = 0; k0 < K; k0 += 16) {
    __builtin_prefetch(arow + k0 + 64, 0, 0);
#pragma unroll
    for (int kk = 0; kk < 16; kk += 4) {
      int k = k0 + kk;
      v2f av;
      av.x = arow[k];
      av.y = arow[k + 1];
      if (aScale) {
        int c0 = k + 2 * half;
        av.x = av.x * aScale[c0] + aShift[c0];
        av.y = av.y * aScale[c0 + 1] + aShift[c0 + 1];
      }
      if (actA) {
        av.x = fmaxf(av.x, 0.0f);
        av.y = fmaxf(av.y, 0.0f);
      }
      const float* b0 = brow + (size_t)(k + 2 * half) * N;
#pragma unroll
      for (int j = 0; j < NT; ++j) {
        v2f bv;
        bv.x = b0[j * 16];
        bv.y = b0[j * 16 + N];
        c[j] = wmma4(av, bv, c[j]);
      }
    }
  }
#pragma unroll
  for (int j = 0; j < NT; ++j) {
    float bias_v = bias ? bias[tn0 + j * 16 + l] : 0.0f;
#pragma unroll
    for (int r = 0; r < 8; ++r) {
      int row = tm + r + 8 * half;
      C[(size_t)row * N + tn0 + j * 16 + l] = c[j][r] + bias_v;
    }
  }
}

// ---------------------------------------------------------------------------
// Batched adjacency GEMM with fused mask + eps epilogue:
//   out[bt,i,:] = sum_j (a[bt,i,j] > thr[bt]) * h[bt,j,:] + eps*h[bt,i,:]
// One wave owns a 16-node x 128-feature strip: adjacency row block streamed
// once; h slice (205 KB per bt) is L2-resident. grid=(4, 512), block=256.
// ---------------------------------------------------------------------------
__global__ void k_adj_gemm(const float* __restrict__ Aadj,
                           const float* __restrict__ thr,
                           const float* __restrict__ Hin,
                           const float* __restrict__ eps_p,
                           float* __restrict__ Out) {
  int bt = blockIdx.y;
  int wave = threadIdx.x >> 5;
  int lane = threadIdx.x & 31;
  int strip = blockIdx.x * 8 + wave;          // node-row tile
  if (strip >= (Nn >> 4)) return;             // 25 strips
  int tm = strip << 4;
  int half = lane >> 4, l = lane & 15;

  float th = thr[bt];
  float eps = eps_p[0];
  const float* arow = Aadj + (size_t)bt * NSLICE + (size_t)(tm + l) * Nn + 2 * half;
  const float* hb = Hin + (size_t)bt * (Nn * Hh);

  v8f c[8];
#pragma unroll
  for (int j = 0; j < 8; ++j) c[j] = (v8f){};

  for (int k0 = 0; k0 < Nn; k0 += 16) {
    __builtin_prefetch(arow + k0 + 64, 0, 0);
#pragma unroll
    for (int kk = 0; kk < 16; kk += 4) {
      int k = k0 + kk;
      v2f av;
      av.x = (arow[k] > th) ? 1.0f : 0.0f;
      av.y = (arow[k + 1] > th) ? 1.0f : 0.0f;
      const float* b0 = hb + (size_t)(k + 2 * half) * Hh + l;
#pragma unroll
      for (int j = 0; j < 8; ++j) {
        v2f bv;
        bv.x = b0[j * 16];
        bv.y = b0[j * 16 + Hh];
        c[j] = wmma4(av, bv, c[j]);
      }
    }
  }
  float* ob = Out + (size_t)bt * (Nn * Hh);
#pragma unroll
  for (int j = 0; j < 8; ++j) {
#pragma unroll
    for (int r = 0; r < 8; ++r) {
      int row = tm + r + 8 * half;
      size_t idx = (size_t)row * Hh + j * 16 + l;
      ob[idx] = c[j][r] + eps * hb[idx];
    }
  }
}

// ---------------------------------------------------------------------------
// Exact 70th percentile per (b,t) slice via 4-pass radix select on the
// monotonic uint mapping of f32, + min-above pass for linear interpolation.
// grid = 512, block = 256.
// ---------------------------------------------------------------------------
static __device__ __forceinline__ unsigned f2key(float f) {
  unsigned bits = __float_as_uint(f);
  return (bits & 0x80000000u) ? ~bits : (bits | 0x80000000u);
}
static __device__ __forceinline__ float key2f(unsigned key) {
  unsigned bits = (key & 0x80000000u) ? (key & 0x7FFFFFFFu) : ~key;
  return __uint_as_float(bits);
}

__global__ void k_percentile(const float* __restrict__ A, float* __restrict__ thr,
                             int n, int k0, float frac) {
  __shared__ unsigned hist[256];
  __shared__ unsigned s_prefix, s_kRem, s_cEq, s_minAbove;
  int bt = blockIdx.x;
  const float* p = A + (size_t)bt * n;
  int tid = threadIdx.x;
  if (tid == 0) { s_prefix = 0u; s_kRem = (unsigned)k0; }
  __syncthreads();
  for (int pass = 0; pass < 4; ++pass) {
    hist[tid] = 0u;
    __syncthreads();
    int shift = 24 - 8 * pass;
    unsigned prefix = s_prefix;
    for (int i = tid; i < n; i += 256) {
      unsigned key = f2key(p[i]);
      if (pass == 0 || (key >> (shift + 8)) == prefix)
        atomicAdd(&hist[(key >> shift) & 0xFFu], 1u);
    }
    __syncthreads();
    if (tid == 0) {
      unsigned kRem = s_kRem, cum = 0, cnt = 0;
      int bsel = 0;
      for (int bb2 = 0; bb2 < 256; ++bb2) {
        cnt = hist[bb2];
        if (cum + cnt > kRem) { bsel = bb2; break; }
        cum += cnt;
      }
      s_kRem = kRem - cum;
      s_prefix = (prefix << 8) | (unsigned)bsel;
      s_cEq = cnt;
    }
    __syncthreads();
  }
  if (tid == 0) s_minAbove = 0xFFFFFFFFu;
  __syncthreads();
  unsigned pfx = s_prefix;
  unsigned localMin = 0xFFFFFFFFu;
  for (int i = tid; i < n; i += 256) {
    unsigned key = f2key(p[i]);
    if (key > pfx && key < localMin) localMin = key;
  }
  atomicMin(&s_minAbove, localMin);
  __syncthreads();
  if (tid == 0) {
    unsigned k1key = (s_kRem + 2u <= s_cEq) ? pfx : s_minAbove;
    float v0 = key2f(pfx);
    float v1 = (k1key == 0xFFFFFFFFu) ? v0 : key2f(k1key);
    thr[bt] = v0 + frac * (v1 - v0);
  }
}

// ---------------------------------------------------------------------------
// Small utility kernels
// ---------------------------------------------------------------------------
__global__ void k_zero(float* __restrict__ p, int n) {
  int i = blockIdx.x * blockDim.x + threadIdx.x;
  if (i < n) p[i] = 0.0f;
}

// Column stats for BatchNorm over M rows x 128 features.
__global__ void k_colstats(const float* __restrict__ X, int M, int rowsPerBlock,
                           float* __restrict__ sums, float* __restrict__ sumsq) {
  int j = threadIdx.x;  // 128
  int r0 = blockIdx.x * rowsPerBlock;
  int r1 = min(r0 + rowsPerBlock, M);
  float s = 0.f, s2 = 0.f;
  for (int r = r0; r < r1; ++r) {
    float x = X[(size_t)r * Hh + j];
    s += x;
    s2 += x * x;
  }
  atomicAdd(&sums[j], s);
  atomicAdd(&sumsq[j], s2);
}

// Per-column BN affine: scale = g*rsqrt(var+eps), shift = be - mean*scale.
__global__ void k_scaleshift(const float* __restrict__ sums,
                             const float* __restrict__ sumsq, float invM,
                             const float* __restrict__ g,
                             const float* __restrict__ be,
                             float* __restrict__ scale,
                             float* __restrict__ shift) {
  int j = threadIdx.x;
  float m = sums[j] * invM;
  float var = sumsq[j] * invM - m * m;   // biased var (ddof=0)
  float sc = g[j] * rsqrtf(var + 1e-5f);
  scale[j] = sc;
  shift[j] = be[j] - m * sc;
}

// In-place BN (+relu or +exact gelu) over [M,128]; act: 0=relu, 1=gelu.
__global__ void k_bn_act(float* __restrict__ X, long long total, float invM,
                         const float* __restrict__ sums,
                         const float* __restrict__ sumsq,
                         const float* __restrict__ g,
                         const float* __restrict__ bb, int act) {
  long long idx = (long long)blockIdx.x * blockDim.x + threadIdx.x;
  if (idx >= total) return;
  int j = (int)(idx & (Hh - 1));
  float m = sums[j] * invM;
  float var = sumsq[j] * invM - m * m;
  float x = X[idx];
  float y = g[j] * (x - m) * rsqrtf(var + 1e-5f) + bb[j];
  if (act == 0)
    y = fmaxf(y, 0.0f);
  else
    y = 0.5f * y * (1.0f + erff(y * 0.70710678118654752f));  // exact gelu
  X[idx] = y;
}

// xr[t*B+b, j] = mean_n h[b,t,n,j]
__global__ void k_node_mean(const float* __restrict__ H, float* __restrict__ xr) {
  int tb = blockIdx.x;
  int t = tb >> 3, b = tb & 7;
  int j = threadIdx.x;
  const float* base = H + (size_t)(b * Tt + t) * Nn * Hh + j;
  float s = 0.f;
  for (int nn = 0; nn < Nn; ++nn) s += base[(size_t)nn * Hh];
  xr[(size_t)tb * Hh + j] = s * (1.0f / Nn);
}

// sigmoid in-place over [512,400]; also scatter to node_attns output [B,L,T,N]
__global__ void k_sigmoid_attn(float* __restrict__ X, float* __restrict__ out,
                               int layer) {
  int idx = blockIdx.x * 256 + threadIdx.x;
  if (idx >= BT * Nn) return;
  int row = idx / Nn;                 // t*B+b
  int n = idx - row * Nn;
  float v = 1.0f / (1.0f + expf(-X[idx]));
  X[idx] = v;
  int t = row >> 3, b = row & 7;
  out[O_NODE + (((size_t)b * Ll + layer) * Tt + t) * Nn + n] = v;
}

// h_read[t*B+b, j] = mean_n h[b,t,n,j] * attn[t*B+b, n]
__global__ void k_wmean(const float* __restrict__ H, const float* __restrict__ attn,
                        float* __restrict__ hread) {
  int tb = blockIdx.x;
  int t = tb >> 3, b = tb & 7;
  int j = threadIdx.x;
  const float* base = H + (size_t)(b * Tt + t) * Nn * Hh + j;
  const float* ar = attn + (size_t)tb * Nn;
  float s = 0.f;
  for (int nn = 0; nn < Nn; ++nn) s += base[(size_t)nn * Hh] * ar[nn];
  hread[(size_t)tb * Hh + j] = s * (1.0f / Nn);
}

// fG[b, layer, j] = mean_t h_read[t*B+b, j]
__global__ void k_fG(const float* __restrict__ hread, float* __restrict__ out,
                     int layer) {
  int b = blockIdx.x, j = threadIdx.x;
  float s = 0.f;
  for (int t = 0; t < Tt; ++t) s += hread[((size_t)t * Bb + b) * Hh + j];
  out[O_FG + ((size_t)b * Ll + layer) * Hh + j] = s * (1.0f / Tt);
}

// Attention scores + softmax: w[b,t,s] = softmax_s(q[t,b]·k[s,b] / sqrt(H))
__global__ void k_scores(const float* __restrict__ Q, const float* __restrict__ K,
                         float* __restrict__ Wbuf, float* __restrict__ out,
                         int layer) {
  int t = blockIdx.x, b = blockIdx.y;
  int s = threadIdx.x;  // 64
  const float* q = Q + ((size_t)t * Bb + b) * Hh;
  const float* kr = K + ((size_t)s * Bb + b) * Hh;
  float acc = 0.f;
  for (int j = 0; j < Hh; ++j) acc += q[j] * kr[j];
  acc *= 0.08838834764831845f;  // 1/sqrt(128)

  __shared__ float red[64];
  red[s] = acc;
  __syncthreads();
  for (int off = 32; off > 0; off >>= 1) {
    if (s < off) red[s] = fmaxf(red[s], red[s + off]);
    __syncthreads();
  }
  float mx = red[0];
  __syncthreads();
  float e = expf(acc - mx);
  red[s] = e;
  __syncthreads();
  for (int off = 32; off > 0; off >>= 1) {
    if (s < off) red[s] += red[s + off];
    __syncthreads();
  }
  float w = e / red[0];
  Wbuf[((size_t)b * Tt + t) * Tt + s] = w;
  out[O_TIME + (((size_t)b * Ll + layer) * Tt + t) * Tt + s] = w;
}

// tmp[t*B+b, j] = sum_s w[b,t,s] * v[s*B+b, j]
__global__ void k_apply(const float* __restrict__ Wbuf, const float* __restrict__ V,
                        float* __restrict__ tmp) {
  int tb = blockIdx.x;
  int t = tb >> 3, b = tb & 7;
  int j = threadIdx.x;
  const float* wr = Wbuf + ((size_t)b * Tt + t) * Tt;
  float acc = 0.f;
  for (int s = 0; s < Tt; ++s) acc += wr[s] * V[((size_t)s * Bb + b) * Hh + j];
  tmp[(size_t)tb * Hh + j] = acc;
}

// LayerNorm over last dim (128): out = g*((A[+Res])-m)*rsqrt(var+eps)+b
__global__ void k_ln(const float* __restrict__ A, const float* __restrict__ Res,
                     const float* __restrict__ g, const float* __restrict__ bb,
                     float* __restrict__ Out) {
  int row = blockIdx.x, j = threadIdx.x;
  float x = A[(size_t)row * Hh + j];
  if (Res) x += Res[(size_t)row * Hh + j];
  __shared__ float s1[Hh], s2[Hh];
  s1[j] = x;
  s2[j] = x * x;
  __syncthreads();
  for (int off = 64; off > 0; off >>= 1) {
    if (j < off) { s1[j] += s1[j + off]; s2[j] += s2[j + off]; }
    __syncthreads();
  }
  float m = s1[0] * (1.0f / Hh);
  float var = s2[0] * (1.0f / Hh) - m * m;
  Out[(size_t)row * Hh + j] = g[j] * (x - m) * rsqrtf(var + 1e-5f) + bb[j];
}

// feat_T[b,j] = sum_t x2[t*B+b, j]  (cls_token='sum')
__global__ void k_featT(const float* __restrict__ X2, float* __restrict__ featT,
                        float* __restrict__ out, int layer) {
  int b = blockIdx.x, j = threadIdx.x;
  float s = 0.f;
  for (int t = 0; t < Tt; ++t) s += X2[((size_t)t * Bb + b) * Hh + j];
  featT[b * Hh + j] = s;
  out[O_FT + ((size_t)b * Ll + layer) * Hh + j] = s;
}

// feat_L = featT @ cls_W + cls_b ; logit += feat_L
__global__ void k_cls(const float* __restrict__ featT, const float* __restrict__ W,
                      const float* __restrict__ bb, float* __restrict__ out,
                      int layer) {
  int tid = threadIdx.x;
  if (tid >= Bb * 2) return;
  int b = tid >> 1, c = tid & 1;
  float acc = bb[c];
  for (int j = 0; j < Hh; ++j) acc += featT[b * Hh + j] * W[j * 2 + c];
  out[O_FL + ((size_t)b * Ll + layer) * 2 + c] = acc;
  out[O_LOGIT + b * 2 + c] += acc;
}

// ---------------------------------------------------------------------------
// Host orchestration
// ---------------------------------------------------------------------------
struct LayerP {
  const float *gin_eps, *gin_W1, *gin_b1, *gin_g1, *gin_be1;
  const float *gin_W2, *gin_b2, *gin_g2, *gin_be2;
  const float *sero_We, *sero_be, *sero_g, *sero_beta, *sero_Wa, *sero_ba;
  const float *Wq, *bq, *Wk, *bk, *Wv, *bv, *Wo, *bo;
  const float *ln1_g, *ln1_b, *ln2_g, *ln2_b;
  const float *mlp_W1, *mlp_b1, *mlp_W2, *mlp_b2;
  const float *cls_W, *cls_b;
};

extern "C" void kernel_launch(void* const* d_in, const int* in_sizes, int n_in,
                              void* d_out, int out_size, void* d_ws, size_t ws_size,
                              hipStream_t stream) {
  (void)in_sizes; (void)n_in; (void)out_size; (void)ws_size;
  const float* v = (const float*)d_in[0];
  const float* a = (const float*)d_in[1];
  // d_in[2] = t (unused), d_in[3] = sampling_endpoints (unused)
  const float* init_W = (const float*)d_in[4];
  const float* init_b = (const float*)d_in[5];

  LayerP lp[Ll];
  for (int l = 0; l < Ll; ++l) {
    int o = 6 + 33 * l;
    auto F = [&](int i) { return (const float*)d_in[o + i]; };
    LayerP& P = lp[l];
    P.gin_eps = F(0);  P.gin_W1 = F(1);  P.gin_b1 = F(2);  P.gin_g1 = F(3);
    P.gin_be1 = F(4);  P.gin_W2 = F(5);  P.gin_b2 = F(6);  P.gin_g2 = F(7);
    P.gin_be2 = F(8);  P.sero_We = F(9); P.sero_be = F(10); P.sero_g = F(11);
    P.sero_beta = F(12); P.sero_Wa = F(13); P.sero_ba = F(14);
    P.Wq = F(15); P.bq = F(16); P.Wk = F(17); P.bk = F(18);
    P.Wv = F(19); P.bv = F(20); P.Wo = F(21); P.bo = F(22);
    P.ln1_g = F(23); P.ln1_b = F(24); P.ln2_g = F(25); P.ln2_b = F(26);
    P.mlp_W1 = F(27); P.mlp_b1 = F(28); P.mlp_W2 = F(29); P.mlp_b2 = F(30);
    P.cls_W = F(31); P.cls_b = F(32);
  }

  float* out = (float*)d_out;

  // Workspace carve-out (256B aligned)
  char* wptr = (char*)d_ws;
  auto alloc = [&](size_t nfloats) {
    float* p = (float*)wptr;
    wptr += ((nfloats * sizeof(float) + 255) / 256) * 256;
    return p;
  };
  const size_t BIG = (size_t)ROWS_BTN * Hh;  // 26,214,400
  float* Hbuf  = alloc(BIG);
  float* Xbuf  = alloc(BIG);
  float* Ybuf  = alloc(BIG);
  float* THR   = alloc(BT);
  float* STATS = alloc(256);            // sums[128] + sumsq[128]
  float* SS    = alloc(256);            // BN fused affine: scale[128]+shift[128]
  float* XR    = alloc((size_t)BT * Hh);
  float* ATTN  = alloc((size_t)BT * Nn);
  float* HREAD = alloc((size_t)BT * Hh);
  float* Qb    = alloc((size_t)BT * Hh);
  float* Kb    = alloc((size_t)BT * Hh);
  float* Vb    = alloc((size_t)BT * Hh);
  float* WATT  = alloc((size_t)Bb * Tt * Tt);
  float* TMP   = alloc((size_t)BT * Hh);
  float* ATTO  = alloc((size_t)BT * Hh);
  float* X1    = alloc((size_t)BT * Hh);
  float* MLP1  = alloc((size_t)BT * 2 * Hh);
  float* MLP2  = alloc((size_t)BT * Hh);
  float* X2b   = alloc((size_t)BT * Hh);
  float* FEATT = alloc((size_t)Bb * Hh);

  auto gemm = [&](const float* A, const float* W, const float* bias, float* C,
                  int M, int N, int K, const float* aSc = nullptr,
                  const float* aSh = nullptr, int actA = 0) {
    int tilesN = N >> 4;
    int nFull = tilesN >> 3;     // 128-wide chunks
    int rem = tilesN & 7;        // remaining 16-wide tiles
    if (nFull) {
      int strips = (M >> 4) * nFull;
      k_gemm_strip<8><<<(strips + 7) / 8, 256, 0, stream>>>(
          A, W, bias, aSc, aSh, actA, C, M, N, K, 0, nFull);
    }
    if (rem) {
      int strips = (M >> 4);
      int blocks = (strips + 7) / 8;
      int tnBase = nFull << 7;
      switch (rem) {
        case 1: k_gemm_strip<1><<<blocks, 256, 0, stream>>>(A, W, bias, aSc, aSh, actA, C, M, N, K, tnBase, 1); break;
        case 2: k_gemm_strip<2><<<blocks, 256, 0, stream>>>(A, W, bias, aSc, aSh, actA, C, M, N, K, tnBase, 1); break;
        case 3: k_gemm_strip<3><<<blocks, 256, 0, stream>>>(A, W, bias, aSc, aSh, actA, C, M, N, K, tnBase, 1); break;
        case 4: k_gemm_strip<4><<<blocks, 256, 0, stream>>>(A, W, bias, aSc, aSh, actA, C, M, N, K, tnBase, 1); break;
        case 5: k_gemm_strip<5><<<blocks, 256, 0, stream>>>(A, W, bias, aSc, aSh, actA, C, M, N, K, tnBase, 1); break;
        case 6: k_gemm_strip<6><<<blocks, 256, 0, stream>>>(A, W, bias, aSc, aSh, actA, C, M, N, K, tnBase, 1); break;
        default: k_gemm_strip<7><<<blocks, 256, 0, stream>>>(A, W, bias, aSc, aSh, actA, C, M, N, K, tnBase, 1); break;
      }
    }
  };
  auto colstats = [&](const float* X, int M, int rowsPerBlock) {
    k_zero<<<1, 256, 0, stream>>>(STATS, 256);
    int sblocks = (M + rowsPerBlock - 1) / rowsPerBlock;
    k_colstats<<<sblocks, Hh, 0, stream>>>(X, M, rowsPerBlock, STATS, STATS + Hh);
  };
  auto bnapply = [&](float* X, int M, const float* g, const float* bb, int act) {
    long long total = (long long)M * Hh;
    int eblocks = (int)((total + 255) / 256);
    k_bn_act<<<eblocks, 256, 0, stream>>>(X, total, 1.0f / (float)M, STATS,
                                          STATS + Hh, g, bb, act);
  };

  // --- logit init ---
  k_zero<<<1, 32, 0, stream>>>(out + O_LOGIT, Bb * 2);

  // --- percentile thresholds: q = 100 - 30 = 70 over 160000 elems ---
  double pos = (double)(NSLICE - 1) * 0.70;
  int k0 = (int)pos;
  float frac = (float)(pos - (double)k0);
  k_percentile<<<BT, 256, 0, stream>>>(a, THR, NSLICE, k0, frac);

  // --- init projection: h = v @ init_W + init_b ---
  gemm(v, init_W, init_b, Hbuf, ROWS_BTN, Hh, Cc);

  float* h = Hbuf;
  float* tA = Xbuf;
  float* tB = Ybuf;

  for (int l = 0; l < Ll; ++l) {
    const LayerP& P = lp[l];

    // --- LayerGIN ---
    k_adj_gemm<<<dim3(4, BT), 256, 0, stream>>>(a, THR, h, P.gin_eps, tA);
    gemm(tA, P.gin_W1, P.gin_b1, tB, ROWS_BTN, Hh, Hh);
    colstats(tB, ROWS_BTN, 256);
    k_scaleshift<<<1, Hh, 0, stream>>>(STATS, STATS + Hh, 1.0f / ROWS_BTN,
                                       P.gin_g1, P.gin_be1, SS, SS + Hh);
    // W2 GEMM with BN1+ReLU fused into A-operand loads
    gemm(tB, P.gin_W2, P.gin_b2, tA, ROWS_BTN, Hh, Hh, SS, SS + Hh, 1);
    colstats(tA, ROWS_BTN, 256);
    bnapply(tA, ROWS_BTN, P.gin_g2, P.gin_be2, /*relu*/ 0);
    { float* oldh = h; h = tA; tA = oldh; }  // h updated, old h is scratch

    // --- ModuleSERO readout ---
    k_node_mean<<<BT, Hh, 0, stream>>>(h, XR);
    gemm(XR, P.sero_We, P.sero_be, TMP, BT, Hh, Hh);
    colstats(TMP, BT, BT);
    bnapply(TMP, BT, P.sero_g, P.sero_beta, /*gelu*/ 1);
    gemm(TMP, P.sero_Wa, P.sero_ba, ATTN, BT, Nn, Hh);
    k_sigmoid_attn<<<(BT * Nn + 255) / 256, 256, 0, stream>>>(ATTN, out, l);
    k_wmean<<<BT, Hh, 0, stream>>>(h, ATTN, HREAD);
    k_fG<<<Bb, Hh, 0, stream>>>(HREAD, out, l);

    // --- ModuleTransformer (1 head) ---
    gemm(HREAD, P.Wq, P.bq, Qb, BT, Hh, Hh);
    gemm(HREAD, P.Wk, P.bk, Kb, BT, Hh, Hh);
    gemm(HREAD, P.Wv, P.bv, Vb, BT, Hh, Hh);
    k_scores<<<dim3(Tt, Bb), Tt, 0, stream>>>(Qb, Kb, WATT, out, l);
    k_apply<<<BT, Hh, 0, stream>>>(WATT, Vb, TMP);
    gemm(TMP, P.Wo, P.bo, ATTO, BT, Hh, Hh);
    k_ln<<<BT, Hh, 0, stream>>>(ATTO, nullptr, P.ln1_g, P.ln1_b, X1);
    gemm(X1, P.mlp_W1, P.mlp_b1, MLP1, BT, 2 * Hh, Hh);
    // second MLP GEMM with ReLU fused into A-operand loads
    gemm(MLP1, P.mlp_W2, P.mlp_b2, MLP2, BT, Hh, 2 * Hh, nullptr, nullptr, 1);
    k_ln<<<BT, Hh, 0, stream>>>(X1, MLP2, P.ln2_g, P.ln2_b, X2b);

    // --- heads ---
    k_featT<<<Bb, Hh, 0, stream>>>(X2b, FEATT, out, l);
    k_cls<<<1, 32, 0, stream>>>(FEATT, P.cls_W, P.cls_b, out, l);
  }
}